// DeformUpSample_67439576481987
// MI455X (gfx1250) — compile-verified
//
#include <hip/hip_runtime.h>
#include <hip/hip_bf16.h>

typedef __attribute__((ext_vector_type(2))) float v2f;
typedef __attribute__((ext_vector_type(8))) float v8f;

#define B_    4
#define CH_   256
#define H_    128
#define W_    128
#define NPIX  (H_ * W_)          // 16384
#define OH_   256
#define OW_   256
#define ONPIX (OH_ * OW_)        // 65536
#define OFFSET_FACTOR 0.70710678118654752440f   // sqrt(128/256)

// ---------------------------------------------------------------------------
// Kernel 1: 1x1 conv (GEMM M=8,K=256,N=16384 per batch) via V_WMMA_F32_16X16X4_F32
// Each wave computes a 16(M, 8 valid)x16(N pixels) tile, K-loop of 64 WMMAs.
// Stores FINAL sampling coordinates (bias + OFFSET_FACTOR + grid folded in):
//   off[b][j   ][p] = xc for subpixel j   (j = a*2+c)
//   off[b][4+j ][p] = yc
// ---------------------------------------------------------------------------
__global__ void deform_conv_offsets_wmma(const float* __restrict__ x,
                                         const float* __restrict__ cw,
                                         const float* __restrict__ cb,
                                         float* __restrict__ off)
{
    const int gwave = (blockIdx.x * blockDim.x + threadIdx.x) >> 5; // 4096 waves
    const int lane  = threadIdx.x & 31;

    const int tilesPerBatch = NPIX / 16;           // 1024
    const int b  = gwave / tilesPerBatch;
    const int pb = (gwave % tilesPerBatch) * 16;   // first pixel of tile

    const int n  = lane & 15;                      // N (pixel) / M (out-ch) slot
    const int kh = lane >> 4;                      // K-half selector

    // A: W[m][k] for m = lane&15 (rows 8..15 discarded), K pair per lane-half
    const float* wrow = cw + (size_t)(n & 7) * CH_;
    // B: X[b][k][pb + n]
    const float* xb = x + (size_t)b * CH_ * NPIX + pb + n;

    v8f acc = {};
    for (int k0 = 0; k0 < CH_; k0 += 4) {
        v2f A, Bf;
        A.x  = wrow[k0 + 2 * kh];
        A.y  = wrow[k0 + 2 * kh + 1];
        Bf.x = xb[(size_t)(k0 + 2 * kh)     * NPIX];
        Bf.y = xb[(size_t)(k0 + 2 * kh + 1) * NPIX];
        acc = __builtin_amdgcn_wmma_f32_16x16x4_f32(
            /*neg_a=*/false, A, /*neg_b=*/false, Bf,
            /*c_mod=*/(short)0, acc, /*reuse_a=*/false, /*reuse_b=*/false);
    }

    // D layout: VGPR v, lanes 0-15 -> M=v (valid channels 0..7); lanes 16-31 -> M=v+8.
    if (lane < 16) {
        const int p    = pb + n;
        const float wc = (float)(p & (W_ - 1));    // input column
        const float hr = (float)(p >> 7);          // input row
        float* ob = off + (size_t)b * 8 * NPIX + p;
#pragma unroll
        for (int v = 0; v < 8; ++v) {
            const float o = OFFSET_FACTOR * (acc[v] + cb[v]);
            float val;
            if (v < 4) {   // x-coordinate channel j=v : + w + (j&1)*0.5
                val = o + wc + (float)(v & 1) * 0.5f;
            } else {       // y-coordinate channel j=v-4 : + h + (j>>1)*0.5
                val = o + hr + (float)((v - 4) >> 1) * 0.5f;
            }
            ob[(size_t)v * NPIX] = val;
        }
    }
}

// ---------------------------------------------------------------------------
// Kernel 2: bilinear gather of all 256 channels at each output pixel.
// Block = one (batch, output row); lane = output column -> coalesced gathers
// that stay L2-resident; output written with non-temporal stores so the
// 268 MB result stream does not evict x (67 MB) from the 192 MB L2.
// ---------------------------------------------------------------------------
__global__ void deform_bilinear_gather(const float* __restrict__ x,
                                       const float* __restrict__ off,
                                       float* __restrict__ out)
{
    const int b  = blockIdx.x >> 8;        // 0..3
    const int oh = blockIdx.x & 255;       // output row
    const int ow = threadIdx.x;            // output column (256 threads)

    const int h = oh >> 1, a = oh & 1;
    const int w = ow >> 1, c = ow & 1;
    const int j = a * 2 + c;
    const int hp = h * W_ + w;

    const float* offb = off + (size_t)b * 8 * NPIX;
    const float xc = offb[(size_t)j * NPIX + hp];
    const float yc = offb[(size_t)(4 + j) * NPIX + hp];

    const float x0f = floorf(xc);
    const float y0f = floorf(yc);
    const float fx = xc - x0f;
    const float fy = yc - y0f;
    const int x0r = (int)x0f;
    const int y0r = (int)y0f;
    const int x0 = min(max(x0r,     0), W_ - 1);
    const int x1 = min(max(x0r + 1, 0), W_ - 1);
    const int y0 = min(max(y0r,     0), H_ - 1);
    const int y1 = min(max(y0r + 1, 0), H_ - 1);

    const float w00 = (1.0f - fx) * (1.0f - fy);
    const float w10 = fx * (1.0f - fy);
    const float w01 = (1.0f - fx) * fy;
    const float w11 = fx * fy;

    const float* xb = x + (size_t)b * CH_ * NPIX;
    const float* p00 = xb + (size_t)(y0 * W_ + x0);
    const float* p10 = xb + (size_t)(y0 * W_ + x1);
    const float* p01 = xb + (size_t)(y1 * W_ + x0);
    const float* p11 = xb + (size_t)(y1 * W_ + x1);

    float* op = out + (size_t)b * CH_ * ONPIX + (size_t)(oh * OW_ + ow);

#pragma unroll 4
    for (int ch = 0; ch < CH_; ++ch) {
        const float v = p00[0] * w00 + p10[0] * w10 + p01[0] * w01 + p11[0] * w11;
        __builtin_nontemporal_store(v, op);
        p00 += NPIX; p10 += NPIX; p01 += NPIX; p11 += NPIX;
        op  += ONPIX;
    }
}

extern "C" void kernel_launch(void* const* d_in, const int* in_sizes, int n_in,
                              void* d_out, int out_size, void* d_ws, size_t ws_size,
                              hipStream_t stream) {
    const float* x  = (const float*)d_in[0];  // (4,256,128,128)
    const float* cw = (const float*)d_in[1];  // (8,256,1,1)
    const float* cb = (const float*)d_in[2];  // (8,)
    float* out = (float*)d_out;               // (4,256,256,256)
    float* off = (float*)d_ws;                // (4,8,16384) = 2 MB scratch

    // Kernel 1: 4096 waves total (4 batches * 1024 tiles), 8 waves/block.
    deform_conv_offsets_wmma<<<dim3(512), dim3(256), 0, stream>>>(x, cw, cb, off);

    // Kernel 2: one block per (batch, output row).
    deform_bilinear_gather<<<dim3(B_ * OH_), dim3(OW_), 0, stream>>>(x, off, out);
}